// SpectrumMatchingLoss_27427661152423
// MI455X (gfx1250) — compile-verified
//
#include <hip/hip_runtime.h>

typedef __attribute__((ext_vector_type(2))) float v2f;
typedef __attribute__((ext_vector_type(8))) float v8f;

#define PROTON_F      1.00782503207f
#define WATER_F       18.0105646863f
#define HUBER_DELTA_F 0.2f

constexpr int Lc = 64;            // sequence length
constexpr int Vc = 28;            // vocab (7 * K=4, exact fit for 16x16x4 WMMA)
constexpr int Mc = 1024;          // observed peaks per batch
constexpr int NPEAK = 2 * (Lc - 1);  // 126 theoretical peaks

// One workgroup (256 threads = 8 waves) per batch element.
// Waves 0-3 compute the 64x28 @ 28 einsum with v_wmma_f32_16x16x4_f32.
__global__ __launch_bounds__(256)
void SpectrumMatchingLoss_kernel(const float* __restrict__ probs,        // B x L x V
                                 const float* __restrict__ obs_mass,     // B x M
                                 const float* __restrict__ obs_inten,    // B x M
                                 const unsigned char* __restrict__ mask, // B x M (bool)
                                 const float* __restrict__ aa,           // V
                                 float* __restrict__ per_batch)          // B
{
    __shared__ float s_aa[Vc];
    __shared__ float s_r[Lc];
    __shared__ float s_prefix[Lc];
    __shared__ float s_theo[NPEAK];
    __shared__ float red_w[256];
    __shared__ float red_i[256];

    const int b    = blockIdx.x;
    const int tid  = threadIdx.x;
    const int lane = tid & 31;
    const int wave = tid >> 5;

    if (tid < Vc) s_aa[tid] = aa[tid];
    __syncthreads();

    // ---- r[l] = sum_v probs[b,l,v] * aa[v] via WMMA f32 16x16x4 ----
    // A tile (16x4): lane m (0-15) holds K=0,1 in the 2 A VGPRs; lane m+16 holds K=2,3.
    // B tile (4x16): aa value broadcast across all 16 columns -> every C column equals r.
    if (wave < 4) {
        const int   m     = lane & 15;
        const int   khigh = lane >> 4;                  // 0 -> K=0,1 ; 1 -> K=2,3
        const float* P = probs + (size_t)b * Lc * Vc + (size_t)(wave * 16 + m) * Vc;
        v8f c = {};
#pragma unroll
        for (int kc = 0; kc < 7; ++kc) {                // 7 * K4 = 28 = V
            const int k0 = 4 * kc + 2 * khigh;
            v2f av; av.x = P[k0];     av.y = P[k0 + 1];
            v2f bv; bv.x = s_aa[k0];  bv.y = s_aa[k0 + 1];
            c = __builtin_amdgcn_wmma_f32_16x16x4_f32(false, av, false, bv,
                                                      (short)0, c, false, false);
        }
        // C layout: VGPR v, lanes 0-15 -> row M=v; lanes 16-31 -> row M=v+8.
        // All columns identical, so lane 0 / lane 16 can write their 8 rows.
        if (m == 0) {
            const int base = wave * 16 + 8 * khigh;
#pragma unroll
            for (int v = 0; v < 8; ++v) s_r[base + v] = c[v];
        }
    }
    __syncthreads();

    // ---- sequential prefix sum (matches jnp.cumsum order) ----
    if (tid == 0) {
        float run = 0.0f;
        for (int l = 0; l < Lc; ++l) { run += s_r[l]; s_prefix[l] = run; }
    }
    __syncthreads();

    // ---- theoretical peaks: 63 b-ions + 63 y-ions ----
    if (tid < Lc - 1) {
        const float total = s_prefix[Lc - 1];
        const float p     = s_prefix[tid];
        s_theo[tid]            = p + PROTON_F;
        s_theo[(Lc - 1) + tid] = total - p + WATER_F + PROTON_F;
    }
    __syncthreads();

    // ---- per-peak min distance, Huber, weighted accumulation ----
    const float* om = obs_mass  + (size_t)b * Mc;
    const float* oi = obs_inten + (size_t)b * Mc;
    const unsigned char* mk = mask + (size_t)b * Mc;

    float wsum = 0.0f, isum = 0.0f;
    for (int i = tid; i < Mc; i += 256) {
        const float o     = om[i];
        const float inten = oi[i];
        float mn = 3.4e38f;
#pragma unroll 2
        for (int j = 0; j < NPEAK; ++j)
            mn = fminf(mn, fabsf(s_theo[j] - o));       // LDS broadcast reads
        const float h = (mn <= HUBER_DELTA_F)
                            ? 0.5f * mn * mn
                            : HUBER_DELTA_F * (mn - 0.5f * HUBER_DELTA_F);
        wsum += h * inten * (mk[i] ? 1.0f : 0.0f);
        isum += inten;                                   // denominator ignores mask
    }
    red_w[tid] = wsum;
    red_i[tid] = isum;
    __syncthreads();

    // deterministic fixed-order tree reduction
    for (int s = 128; s > 0; s >>= 1) {
        if (tid < s) { red_w[tid] += red_w[tid + s]; red_i[tid] += red_i[tid + s]; }
        __syncthreads();
    }
    if (tid == 0)
        per_batch[b] = red_w[0] / fmaxf(red_i[0], 1e-8f);   // clip(sum, 1e-8)
}

__global__ __launch_bounds__(256)
void SpectrumMatchingLoss_mean_kernel(const float* __restrict__ per_batch,
                                      float* __restrict__ out, int B)
{
    __shared__ float red[256];
    float s = 0.0f;
    for (int i = threadIdx.x; i < B; i += 256) s += per_batch[i];
    red[threadIdx.x] = s;
    __syncthreads();
    for (int st = 128; st > 0; st >>= 1) {
        if (threadIdx.x < st) red[threadIdx.x] += red[threadIdx.x + st];
        __syncthreads();
    }
    if (threadIdx.x == 0) out[0] = red[0] / (float)B;
}

extern "C" void kernel_launch(void* const* d_in, const int* in_sizes, int n_in,
                              void* d_out, int out_size, void* d_ws, size_t ws_size,
                              hipStream_t stream) {
    const float*         probs = (const float*)d_in[0];         // (B, 64, 28) f32
    const float*         om    = (const float*)d_in[1];         // (B, 1024) f32
    const float*         oi    = (const float*)d_in[2];         // (B, 1024) f32
    const unsigned char* mk    = (const unsigned char*)d_in[3]; // (B, 1024) bool
    const float*         aa    = (const float*)d_in[4];         // (28,) f32

    const int B = in_sizes[1] / Mc;   // 512
    float* per_batch = (float*)d_ws;  // B floats of scratch, fully rewritten each call

    SpectrumMatchingLoss_kernel<<<B, 256, 0, stream>>>(probs, om, oi, mk, aa, per_batch);
    SpectrumMatchingLoss_mean_kernel<<<1, 256, 0, stream>>>(per_batch, (float*)d_out, B);
}